// knn_euclidean_64493228917427
// MI455X (gfx1250) — compile-verified
//
#include <hip/hip_runtime.h>
#include <hip/hip_bf16.h>

// ---------------------------------------------------------------------------
// KNN (K=16, threshold 50) over coords[2, 8192, 256] fp32 on gfx1250.
// Fused tiled X·X^T (V_WMMA_F32_16X16X4_F32) + streaming per-row top-16.
// 4 independent WMMA accumulator chains to keep the XDL pipeline full.
// ---------------------------------------------------------------------------

typedef __attribute__((ext_vector_type(2))) float v2f;
typedef __attribute__((ext_vector_type(4))) float v4f;
typedef __attribute__((ext_vector_type(8))) float v8f;

#define BATCH     2
#define NPTS      8192
#define DIMS      256
#define KNN       16
#define THRESH    50.0f
#define ROWS_WG   32          // rows of the distance matrix per workgroup (2 M-tiles)
#define LDSTRIDE  260         // padded floats/row: 260 % 64 == 4 -> conflict-free b64 frags

// --------------------------- kernel 1: row norms ---------------------------
__global__ __launch_bounds__(256) void xsq_kernel(const float* __restrict__ X,
                                                  float* __restrict__ xsq) {
    const int gwave = (blockIdx.x * 256 + threadIdx.x) >> 5;   // one wave per row
    const int lane  = threadIdx.x & 31;
    if (gwave >= BATCH * NPTS) return;
    const float* row = X + (size_t)gwave * DIMS;
    v4f a = *(const v4f*)(row + lane * 4);          // cols [4*lane, 4*lane+3]
    v4f b = *(const v4f*)(row + 128 + lane * 4);    // cols [128+4*lane, ...]
    float s = a[0]*a[0] + a[1]*a[1] + a[2]*a[2] + a[3]*a[3]
            + b[0]*b[0] + b[1]*b[1] + b[2]*b[2] + b[3]*b[3];
    #pragma unroll
    for (int off = 16; off > 0; off >>= 1) s += __shfl_xor(s, off, 32);
    if (lane == 0) xsq[gwave] = s;
}

// --------------------------- kernel 2: fused knn ---------------------------
__global__ __launch_bounds__(32) void knn_main(const float* __restrict__ X,
                                               const float* __restrict__ xsq,
                                               int* __restrict__ out) {
    __shared__ float a_lds[ROWS_WG * LDSTRIDE];   // 32 rows of A, padded
    __shared__ float b_lds[16 * LDSTRIDE];        // 16-row j-tile, padded
    __shared__ float dist_lds[ROWS_WG * 17];      // transpose buffer, padded

    const int wg   = blockIdx.x;                  // 0..511
    const int b    = wg >> 8;                     // 256 row-groups per batch
    const int i0   = (wg & 255) * ROWS_WG;
    const int lane = threadIdx.x;

    const float* Xb = X + (size_t)b * NPTS * DIMS;

    // Stage A: 32 rows x 256 floats, coalesced float4 (2048 float4 / 32 lanes)
    for (int c = 0; c < 64; ++c) {
        const int f    = c * 32 + lane;           // flat float4 index
        const int row  = f >> 6;                  // 64 float4 per row
        const int col4 = f & 63;
        v4f v = *(const v4f*)(Xb + (size_t)(i0 + row) * DIMS + col4 * 4);
        *(v4f*)(&a_lds[row * LDSTRIDE + col4 * 4]) = v;
    }

    const float xsq_own = xsq[b * NPTS + i0 + lane];  // lane owns row i0+lane

    float best_d[KNN];
    int   best_i[KNN];
    #pragma unroll
    for (int k = 0; k < KNN; ++k) { best_d[k] = __builtin_inff(); best_i[k] = k; }

    // f32 16x16x4 fragment addressing: lane m = L%16; K = {0,1} (lanes 0-15),
    // {2,3} (lanes 16-31); one float2 per VGPR-pair slice.
    const int mrow  = lane & 15;
    const int khalf = (lane >> 4) * 2;

    __syncthreads();

    for (int j0 = 0; j0 < NPTS; j0 += 16) {
        // Stage B tile: 8 rows x 256 floats, coalesced (512 float4 / 32 lanes)
        for (int c = 0; c < 16; ++c) {
            const int f    = c * 32 + lane;
            const int row  = f >> 6;
            const int col4 = f & 63;
            v4f v = *(const v4f*)(Xb + (size_t)(j0 + row) * DIMS + col4 * 4);
            *(v4f*)(&b_lds[row * LDSTRIDE + col4 * 4]) = v;
        }
        // per-lane ||x_j||^2 for column (lane & 15) of this tile
        const float xsq_j = xsq[b * NPTS + j0 + mrow];

        // prefetch next tile into cache while we compute
        if (j0 + 16 < NPTS)
            __builtin_prefetch(Xb + (size_t)(j0 + 16) * DIMS + lane * 32, 0, 1);

        __syncthreads();

        // 4 independent accumulator chains: {M-tile 0/1} x {even/odd k-step}
        v8f acc0e = {}, acc0o = {};   // rows i0..i0+15
        v8f acc1e = {}, acc1o = {};   // rows i0+16..i0+31
        #pragma unroll 8
        for (int t = 0; t < 64; t += 2) {
            const int colE = 4 * t + khalf;
            const int colO = colE + 4;
            v2f bfE = *(const v2f*)(&b_lds[mrow * LDSTRIDE + colE]);
            v2f a0E = *(const v2f*)(&a_lds[mrow * LDSTRIDE + colE]);
            v2f a1E = *(const v2f*)(&a_lds[(16 + mrow) * LDSTRIDE + colE]);
            v2f bfO = *(const v2f*)(&b_lds[mrow * LDSTRIDE + colO]);
            v2f a0O = *(const v2f*)(&a_lds[mrow * LDSTRIDE + colO]);
            v2f a1O = *(const v2f*)(&a_lds[(16 + mrow) * LDSTRIDE + colO]);
            acc0e = __builtin_amdgcn_wmma_f32_16x16x4_f32(
                        false, a0E, false, bfE, (short)0, acc0e, false, false);
            acc1e = __builtin_amdgcn_wmma_f32_16x16x4_f32(
                        false, a1E, false, bfE, (short)0, acc1e, false, false);
            acc0o = __builtin_amdgcn_wmma_f32_16x16x4_f32(
                        false, a0O, false, bfO, (short)0, acc0o, false, false);
            acc1o = __builtin_amdgcn_wmma_f32_16x16x4_f32(
                        false, a1O, false, bfO, (short)0, acc1o, false, false);
        }

        // C layout: lane holds col n = lane%16, rows m = v + 8*(lane/16).
        // Store xsq_j - 2*inner; lane adds its own xsq at scan time.
        #pragma unroll
        for (int v = 0; v < 8; ++v) {
            const int m = (lane >> 4) * 8 + v;
            const float in0 = acc0e[v] + acc0o[v];
            const float in1 = acc1e[v] + acc1o[v];
            dist_lds[m * 17 + mrow]        = xsq_j - 2.0f * in0;
            dist_lds[(16 + m) * 17 + mrow] = xsq_j - 2.0f * in1;
        }
        __syncthreads();

        // Streaming top-16 insert: lane owns row (i0 + lane); scan 16 candidates
        #pragma unroll 4
        for (int n = 0; n < 16; ++n) {
            const float d = xsq_own + dist_lds[lane * 17 + n];
            if (d <= THRESH && d < best_d[KNN - 1]) {
                best_d[KNN - 1] = d;
                best_i[KNN - 1] = j0 + n;
                #pragma unroll
                for (int t = KNN - 1; t > 0; --t) {
                    if (best_d[t] < best_d[t - 1]) {   // strict: stable ties -> lower idx first
                        float td = best_d[t]; best_d[t] = best_d[t - 1]; best_d[t - 1] = td;
                        int   ti = best_i[t]; best_i[t] = best_i[t - 1]; best_i[t - 1] = ti;
                    }
                }
            }
        }
        __syncthreads();
    }

    // edge_index [2, B, N, K]: slab 0 = nn_idx, slab 1 = center idx
    const int i = i0 + lane;
    const size_t base0 = ((size_t)(0 * BATCH + b) * NPTS + i) * KNN;
    const size_t base1 = ((size_t)(1 * BATCH + b) * NPTS + i) * KNN;
    #pragma unroll
    for (int k = 0; k < KNN; ++k) {
        out[base0 + k] = best_i[k];
        out[base1 + k] = i;
    }
}

// ------------------------------- launcher ----------------------------------
extern "C" void kernel_launch(void* const* d_in, const int* in_sizes, int n_in,
                              void* d_out, int out_size, void* d_ws, size_t ws_size,
                              hipStream_t stream) {
    const float* coords = (const float*)d_in[0];
    int*   out = (int*)d_out;
    float* xsq = (float*)d_ws;   // needs 2*8192*4 = 64 KB of scratch

    // kernel 1: row squared norms (one wave32 per row)
    const int nrows  = BATCH * NPTS;                 // 16384 waves
    const int blocks = (nrows * 32 + 255) / 256;     // 2048 blocks of 8 waves
    xsq_kernel<<<blocks, 256, 0, stream>>>(coords, xsq);

    // kernel 2: fused gram + top-k, 512 single-wave workgroups
    knn_main<<<BATCH * (NPTS / ROWS_WG), 32, 0, stream>>>(coords, xsq, out);
}